// SGC_60395830117192
// MI455X (gfx1250) — compile-verified
//
#include <hip/hip_runtime.h>

typedef __attribute__((ext_vector_type(2))) float v2f;
typedef __attribute__((ext_vector_type(8))) float v8f;

#define IN_DIM 256
#define EMB    128

// No-return fp32 global atomic add, emitted directly as the CDNA5 ISA
// instruction (GLOBAL_ATOMIC_ADD_F32, no-return form -> STOREcnt path,
// RMW executes in L2's atomic units). Device scope for cross-workgroup
// accumulation.
__device__ __forceinline__ void gfx1250_atomic_fadd(float* p, float v)
{
    asm volatile("global_atomic_add_f32 %0, %1, off scope:SCOPE_DEV"
                 :
                 : "v"(p), "v"(v)
                 : "memory");
}

// ---------------------------------------------------------------------------
// Propagation step:  hout[dst] += hin[src] * w   (COO scatter)
// One wave32 per edge; lane l owns features [4l, 4l+4):
//   - edge metadata loads forced scalar (readfirstlane -> SGPR path)
//   - gather  = one global_load_b128 per lane (512B/edge, fully coalesced)
//   - scatter = 4 no-return fp32 hw atomics per lane (h arrays are
//     L2-resident: 51MB each vs MI455X's 192MB L2)
// ---------------------------------------------------------------------------
__global__ __launch_bounds__(256) void sgc_spmm_scatter(
    const float* __restrict__ hin,   // [N, 128]
    float* __restrict__ hout,        // [N, 128]
    const int* __restrict__ esrc,
    const int* __restrict__ edst,
    const float* __restrict__ ew,
    int E)
{
    long t   = (long)blockIdx.x * blockDim.x + threadIdx.x;
    int lane = (int)(t & 31);
    int e    = (int)(t >> 5);
    // e is uniform across the wave (blockDim=256 = 8 full wave32s); make it
    // architecturally uniform so edge metadata goes through the scalar path.
    e = __builtin_amdgcn_readfirstlane(e);
    if (e >= E) return;

    const int   s  = esrc[e];
    const int   d  = edst[e];
    const float wt = ew[e];

    float4 v = ((const float4*)(hin + (size_t)s * EMB))[lane];
    float* ho = hout + (size_t)d * EMB + lane * 4;

    gfx1250_atomic_fadd(ho + 0, v.x * wt);
    gfx1250_atomic_fadd(ho + 1, v.y * wt);
    gfx1250_atomic_fadd(ho + 2, v.z * wt);
    gfx1250_atomic_fadd(ho + 3, v.w * wt);
}

// ---------------------------------------------------------------------------
// Zero-fill (float4 granularity)
// ---------------------------------------------------------------------------
__global__ __launch_bounds__(256) void sgc_zero_f4(float4* __restrict__ p, size_t n4)
{
    size_t i = (size_t)blockIdx.x * blockDim.x + threadIdx.x;
    if (i < n4) p[i] = make_float4(0.f, 0.f, 0.f, 0.f);
}

// ---------------------------------------------------------------------------
// h = relu(x @ W + b)
// One block = 256 threads = 8 wave32.  Block computes a 16-row x 128-col slab.
// Wave w computes the 16x16 tile at columns [16w, 16w+16) with
// V_WMMA_F32_16X16X4_F32 (fp32 in / fp32 accumulate, matches fp32 reference).
// ---------------------------------------------------------------------------
__global__ __launch_bounds__(256) void sgc_gemm_relu_wmma(
    const float* __restrict__ x,     // [N, 256]
    const float* __restrict__ W,     // [256, 128]
    const float* __restrict__ bias,  // [128]
    float* __restrict__ out,         // [N, 128]
    int nrows)
{
    // 16 rows x 256 cols of x, padded to 260 floats/row so A-fragment reads
    // (stride = one row) hit distinct LDS banks (lid*260 % 64 = lid*4).
    __shared__ float xs[16][IN_DIM + 4];

    const int tid  = threadIdx.x;
    const int row0 = blockIdx.x * 16;

    // Cooperative, fully coalesced 16KB slab load (block-uniform control flow).
#pragma unroll
    for (int i = 0; i < 16; ++i) {
        int r = row0 + i;
        if (r >= nrows) r = nrows - 1;           // block-uniform clamp
        xs[i][tid] = x[(size_t)r * IN_DIM + tid];
    }
    __syncthreads();

    const int wave = tid >> 5;
    const int lane = tid & 31;
    const int half = lane >> 4;   // 0: lanes 0-15, 1: lanes 16-31
    const int lid  = lane & 15;
    const int col  = wave * 16 + lid;

    v8f acc = {};

    // A-frag (16x4 f32): lanes 0-15 -> M=lid, K={k0,k0+1};
    //                    lanes 16-31 -> M=lid, K={k0+2,k0+3}.
    const float2* arow = (const float2*)(&xs[lid][0]);

#pragma unroll 8
    for (int k0 = 0; k0 < IN_DIM; k0 += 4) {
        float2 a2 = arow[(k0 >> 1) + half];
        v2f a; a.x = a2.x; a.y = a2.y;

        // B-frag (4x16 f32): lanes index N=col; half selects the K pair.
        const int kb = k0 + 2 * half;
        v2f b;
        b.x = W[(size_t)kb       * EMB + col];
        b.y = W[(size_t)(kb + 1) * EMB + col];

        acc = __builtin_amdgcn_wmma_f32_16x16x4_f32(
            /*neg_a=*/false, a, /*neg_b=*/false, b,
            /*c_mod=*/(short)0, acc, /*reuse_a=*/false, /*reuse_b=*/false);
    }

    const float bv = bias[col];
    // C/D layout: VGPR i -> row M = half*8 + i, col N = lane&15.
    float* obase = out + (size_t)(row0 + half * 8) * EMB + col;

    if (row0 + 16 <= nrows) {
        // Full tile (always true when N % 16 == 0): branchless stores with
        // immediate offsets off one base address.
#pragma unroll
        for (int i = 0; i < 8; ++i) {
            float v = acc[i] + bv;
            obase[(size_t)i * EMB] = v > 0.f ? v : 0.f;
        }
    } else {
#pragma unroll
        for (int i = 0; i < 8; ++i) {
            int row = row0 + half * 8 + i;
            if (row < nrows) {
                float v = acc[i] + bv;
                out[(size_t)row * EMB + col] = v > 0.f ? v : 0.f;
            }
        }
    }
}

// ---------------------------------------------------------------------------
// Launch: gemm -> d_out;  zero ws;  scatter(d_out -> ws);
//         zero d_out;  scatter(ws -> d_out).
// Only N*128*4 = 51.2MB of workspace needed.
// ---------------------------------------------------------------------------
extern "C" void kernel_launch(void* const* d_in, const int* in_sizes, int n_in,
                              void* d_out, int out_size, void* d_ws, size_t ws_size,
                              hipStream_t stream)
{
    const float* x    = (const float*)d_in[0];
    const float* W    = (const float*)d_in[1];
    const float* bias = (const float*)d_in[2];
    const int*   esrc = (const int*)d_in[3];
    const int*   edst = (const int*)d_in[4];
    const float* ew   = (const float*)d_in[5];

    const int N = in_sizes[0] / IN_DIM;
    const int E = in_sizes[3];

    float* out = (float*)d_out;
    float* h1  = (float*)d_ws;     // intermediate [N, 128]

    const int    gemm_blocks = (N + 15) / 16;
    const size_t n4          = (size_t)N * EMB / 4;
    const int    zero_blocks = (int)((n4 + 255) / 256);
    const long   sc_threads  = (long)E * 32;
    const int    sc_blocks   = (int)((sc_threads + 255) / 256);

    // h0 = relu(x @ W + b)  -> d_out
    sgc_gemm_relu_wmma<<<gemm_blocks, 256, 0, stream>>>(x, W, bias, out, N);

    // h1 = A @ h0
    sgc_zero_f4<<<zero_blocks, 256, 0, stream>>>((float4*)h1, n4);
    sgc_spmm_scatter<<<sc_blocks, 256, 0, stream>>>(out, h1, esrc, edst, ew, E);

    // h2 = A @ h1 -> d_out
    sgc_zero_f4<<<zero_blocks, 256, 0, stream>>>((float4*)out, n4);
    sgc_spmm_scatter<<<sc_blocks, 256, 0, stream>>>(h1, out, esrc, edst, ew, E);
}